// SpikingLayer_781684048099
// MI455X (gfx1250) — compile-verified
//
#include <hip/hip_runtime.h>

// Spiking layer step on MI455X (gfx1250):
//   x = W @ f  (W symmetric => x[n] = sum_k W[k,n] f[k], row-major coalesced)
//   v' = 0.9 v + x ; fire = v' > 1 ; v_out = fire ? 0 : v'
// Memory-bound: ~1 GiB of W streamed once (NT hints), WMMA f32 16x16x4 does
// the dot products for free.

typedef __attribute__((ext_vector_type(2)))  float v2f;
typedef __attribute__((ext_vector_type(8)))  float v8f;
typedef __attribute__((ext_vector_type(16))) float v16f;

#define NN      16384
#define TILES   2          // 16-column tiles per block
#define KSLICES 4          // K-split per block (one wave per tile x slice)
#define LEAKF   0.9f
#define THRESHF 1.0f

__global__ __launch_bounds__(256)
void spiking_gemv_wmma(const float* __restrict__ W,
                       const float* __restrict__ f,
                       const float* __restrict__ v,
                       float* __restrict__ out)
{
    __shared__ float red[TILES * KSLICES * 16];

    const int lane = threadIdx.x & 31;
    const int hi   = lane >> 4;        // half-wave: 0 -> lanes 0-15, 1 -> 16-31
    const int ii   = lane & 15;
    const int wave = __builtin_amdgcn_readfirstlane((int)(threadIdx.x >> 5));
    const int tile  = wave & (TILES - 1);
    const int slice = wave >> 1;

    const int col = blockIdx.x * (TILES * 16) + tile * 16 + ii;
    const int ks  = slice * (NN / KSLICES);
    const int ke  = ks + (NN / KSLICES);

    // Per-lane row offsets (relative to W + k0*NN) implementing the B-operand
    // layout of V_WMMA_F32_16X16X4_F32: V0 holds K0 (lanes 0-15) | K2 (16-31),
    // V1 holds K1 | K3 — same half-split slot map as the documented A layout.
    const int r0 = (hi ? 2 : 0) * NN + col;
    const int r1 = (hi ? 3 : 1) * NN + col;

    v8f c[4];
    c[0] = (v8f){}; c[1] = (v8f){}; c[2] = (v8f){}; c[3] = (v8f){};

    for (int k0 = ks; k0 < ke; k0 += 16) {
        // Wave-uniform spike values: one scalar s_load_b512, reused by 64 MACs.
        const v16f fk = *(const v16f*)(f + k0);
        const float* Wk = W + (size_t)k0 * NN;
#pragma unroll
        for (int j = 0; j < 4; ++j) {
            const float* Wkj = Wk + (size_t)(4 * j) * NN;
            v2f a, b;
            a.x = hi ? fk[4 * j + 2] : fk[4 * j + 0];
            a.y = hi ? fk[4 * j + 3] : fk[4 * j + 1];
            // Streaming W: non-temporal, each byte read exactly once chip-wide.
            b.x = __builtin_nontemporal_load(Wkj + r0);
            b.y = __builtin_nontemporal_load(Wkj + r1);
            c[j] = __builtin_amdgcn_wmma_f32_16x16x4_f32(
                       false, a, false, b, (short)0, c[j], false, false);
        }
    }

    // D rows are all identical (A was a broadcast vector): lane ii of each
    // half-wave holds the partial x for column col in every C VGPR; take [0].
    const v8f cs = c[0] + c[1] + c[2] + c[3];
    const float xpart = cs[0];

    if (lane < 16)
        red[tile * (KSLICES * 16) + slice * 16 + ii] = xpart;
    __syncthreads();

    // One thread per output column: reduce K-slices + fused LIF update.
    if (threadIdx.x < TILES * 16) {
        const int l  = threadIdx.x;
        const int t  = l >> 4;
        const int i2 = l & 15;
        float x = 0.0f;
#pragma unroll
        for (int s = 0; s < KSLICES; ++s)
            x += red[t * (KSLICES * 16) + s * 16 + i2];
        const int oc = blockIdx.x * (TILES * 16) + l;
        const float vn = LEAKF * v[oc] + x;
        const bool fire = vn > THRESHF;
        out[oc]      = fire ? 1.0f : 0.0f;   // firing
        out[NN + oc] = fire ? 0.0f : vn;     // v_out
    }
}

extern "C" void kernel_launch(void* const* d_in, const int* in_sizes, int n_in,
                              void* d_out, int out_size, void* d_ws, size_t ws_size,
                              hipStream_t stream) {
    const float* W = (const float*)d_in[0];   // [N, N] f32
    const float* f = (const float*)d_in[1];   // [N] f32 spikes
    const float* v = (const float*)d_in[2];   // [N] f32 potentials
    float* out = (float*)d_out;               // [2N] f32: firing ++ v_out

    dim3 grid(NN / (TILES * 16));             // 512 blocks
    dim3 block(256);                          // 8 wave32s
    hipLaunchKernelGGL(spiking_gemv_wmma, grid, block, 0, stream, W, f, v, out);
}